// TSNE_4990751998273
// MI455X (gfx1250) — compile-verified
//
#include <hip/hip_runtime.h>

typedef float v2f __attribute__((ext_vector_type(2)));
typedef float v8f __attribute__((ext_vector_type(8)));

#define N_POINTS 8192
#define N_PAIRS  262144
#define TILES    (N_POINTS / 16)   // 512 tiles of 16 points

// ws layout (floats): [0]=S1, [1]=Sp, [2]=denom_sum

__global__ void tsne_init_ws(float* ws) {
    if (threadIdx.x < 4) ws[threadIdx.x] = 0.0f;
}

// ----- pair term: grid-stride gather over 262144 sampled pairs --------------
__global__ void tsne_pair_kernel(const float* __restrict__ x,
                                 const float* __restrict__ pij,
                                 const long long* __restrict__ ii,
                                 const long long* __restrict__ jj,
                                 float* __restrict__ ws, int P) {
    __shared__ float s1[8], s2[8];
    int tid = blockIdx.x * blockDim.x + threadIdx.x;
    int stride = gridDim.x * blockDim.x;
    float t1 = 0.f, t2 = 0.f;
    for (int p = tid; p < P; p += stride) {
        long long a = ii[p];
        long long b = jj[p];
        float2 xa = ((const float2*)x)[a];
        float2 xb = ((const float2*)x)[b];
        float dx = xa.x - xb.x, dy = xa.y - xb.y;
        float d2 = __builtin_fmaf(dx, dx, dy * dy);
        float pv = pij[p];
        t1 += pv * (__logf(pv) + __logf(1.0f + d2));
        t2 += pv;
    }
    for (int off = 16; off > 0; off >>= 1) {
        t1 += __shfl_down(t1, off, 32);
        t2 += __shfl_down(t2, off, 32);
    }
    int lane = threadIdx.x & 31, wv = threadIdx.x >> 5;
    if (lane == 0) { s1[wv] = t1; s2[wv] = t2; }
    __syncthreads();
    if (wv == 0) {
        int nw = blockDim.x >> 5;
        t1 = (lane < nw) ? s1[lane] : 0.f;
        t2 = (lane < nw) ? s2[lane] : 0.f;
        for (int off = 4; off > 0; off >>= 1) {
            t1 += __shfl_down(t1, off, 32);
            t2 += __shfl_down(t2, off, 32);
        }
        if (lane == 0) { atomicAdd(&ws[0], t1); atomicAdd(&ws[1], t2); }
    }
}

// ----- denominator via WMMA with homogeneous coordinates --------------------
// A row r = (-2*xr0, -2*xr1, sq_r+1, 1); B col c = (xc0, xc1, 1, sq_c)
//   => WMMA output element = 1 + d2(r,c) directly; epilogue = rcp + add only.
// Operand build per wave32 f32 layout: lanes 0-15 hold K0/K1, lanes 16-31 K2/K3.
__device__ __forceinline__ v2f make_a(const float* x, int tile, int c, bool lo) {
    float2 p = ((const float2*)x)[tile * 16 + c];
    float sq = __builtin_fmaf(p.x, p.x, p.y * p.y);
    v2f a;
    a.x = lo ? -2.0f * p.x : sq + 1.0f;   // K0 | K2
    a.y = lo ? -2.0f * p.y : 1.0f;        // K1 | K3
    return a;
}

__device__ __forceinline__ v2f make_b(const float* x, int tile, int c, bool lo) {
    float2 q = ((const float2*)x)[tile * 16 + c];
    float sq = __builtin_fmaf(q.x, q.x, q.y * q.y);
    v2f b;
    b.x = lo ? q.x : 1.0f;                // K0 | K2
    b.y = lo ? q.y : sq;                  // K1 | K3
    return b;
}

__global__ void tsne_denom_kernel(const float* __restrict__ x,
                                  float* __restrict__ ws) {
    int lane = threadIdx.x & 31;
    int wv   = threadIdx.x >> 5;
    int nw   = blockDim.x >> 5;          // 4 waves/block
    int c    = lane & 15;
    bool lo  = lane < 16;

    v2f a = make_a(x, blockIdx.x, c, lo);   // fixed I-tile per wave

    float acc0 = 0.f, acc1 = 0.f, acc2 = 0.f, acc3 = 0.f;
    // TILES=512, nw=4: each wave does 128 tiles; unroll 2 -> two WMMAs in flight
    for (int J = wv; J < TILES; J += 2 * nw) {
        v2f b0 = make_b(x, J,      c, lo);
        v2f b1 = make_b(x, J + nw, c, lo);

        v8f g0 = {0.f, 0.f, 0.f, 0.f, 0.f, 0.f, 0.f, 0.f};
        v8f g1 = {0.f, 0.f, 0.f, 0.f, 0.f, 0.f, 0.f, 0.f};
        g0 = __builtin_amdgcn_wmma_f32_16x16x4_f32(
                 false, a, false, b0, (short)0, g0, false, false);
        g1 = __builtin_amdgcn_wmma_f32_16x16x4_f32(
                 false, a, false, b1, (short)0, g1, false, false);

        // g = 1 + d2 for all 256 pairs of each tile; just rcp + accumulate
        acc0 += __builtin_amdgcn_rcpf(g0[0]) + __builtin_amdgcn_rcpf(g0[4]);
        acc1 += __builtin_amdgcn_rcpf(g0[1]) + __builtin_amdgcn_rcpf(g0[5]);
        acc2 += __builtin_amdgcn_rcpf(g0[2]) + __builtin_amdgcn_rcpf(g0[6]);
        acc3 += __builtin_amdgcn_rcpf(g0[3]) + __builtin_amdgcn_rcpf(g0[7]);
        acc0 += __builtin_amdgcn_rcpf(g1[0]) + __builtin_amdgcn_rcpf(g1[4]);
        acc1 += __builtin_amdgcn_rcpf(g1[1]) + __builtin_amdgcn_rcpf(g1[5]);
        acc2 += __builtin_amdgcn_rcpf(g1[2]) + __builtin_amdgcn_rcpf(g1[6]);
        acc3 += __builtin_amdgcn_rcpf(g1[3]) + __builtin_amdgcn_rcpf(g1[7]);
    }
    float acc = (acc0 + acc1) + (acc2 + acc3);

    for (int off = 16; off > 0; off >>= 1) acc += __shfl_down(acc, off, 32);
    if (lane == 0) atomicAdd(&ws[2], acc);
}

// ----- final combine: loss = S1 + Sp * log(denom_sum - P) -------------------
__global__ void tsne_combine(const float* __restrict__ ws,
                             float* __restrict__ out) {
    float denom = ws[2] - (float)N_PAIRS;
    out[0] = ws[0] + ws[1] * __logf(denom);
}

extern "C" void kernel_launch(void* const* d_in, const int* in_sizes, int n_in,
                              void* d_out, int out_size, void* d_ws, size_t ws_size,
                              hipStream_t stream) {
    const float*     x   = (const float*)d_in[0];      // [8192, 2] f32
    const float*     pij = (const float*)d_in[1];      // [262144] f32
    const long long* ii  = (const long long*)d_in[2];  // [262144] i64
    const long long* jj  = (const long long*)d_in[3];  // [262144] i64
    float* ws  = (float*)d_ws;
    float* out = (float*)d_out;

    tsne_init_ws<<<1, 32, 0, stream>>>(ws);
    tsne_pair_kernel<<<512, 256, 0, stream>>>(x, pij, ii, jj, ws, N_PAIRS);
    tsne_denom_kernel<<<TILES, 128, 0, stream>>>(x, ws);   // 512 blocks x 4 waves
    tsne_combine<<<1, 1, 0, stream>>>(ws, out);
}